// ValueIteration_59622736003232
// MI455X (gfx1250) — compile-verified
//
#include <hip/hip_runtime.h>

typedef __attribute__((ext_vector_type(2))) float v2f;
typedef __attribute__((ext_vector_type(8))) float v8f;

#define BATCH 16
#define HH    128
#define WW    128
#define RDIM  4
#define IDIM  5
#define ADIM  8
#define KSQ   9
#define IKK   45          // I*K*K logits per action / patch length
#define KPAD  48          // K padded to 12 wmma k-steps of 4
#define MROWS 368         // weight rows staged in LDS (covers c up to 362, padded)
#define TILE  16          // pixels per wave (WMMA N)
#define WAVES 8
#define BLOCK 256

#define WLDS_FLOATS (MROWS * KPAD)     // 17664 (= 69 * 256 exactly)
#define PLDS_FLOATS (KPAD * TILE)      // 768 per wave
#define LDS_BYTES   ((WLDS_FLOATS + WAVES * PLDS_FLOATS) * 4)  // 95232 B

__global__ __launch_bounds__(BLOCK)
void vi_wmma_kernel(const float* __restrict__ values,
                    const float* __restrict__ rewards,
                    const float* __restrict__ weight,
                    float* __restrict__ out)
{
    extern __shared__ float smem[];
    float* wlds = smem;                                   // [368][48] padded weights
    const int tid  = threadIdx.x;
    const int lane = tid & 31;
    const int wave = tid >> 5;
    float* plds = smem + WLDS_FLOATS + wave * PLDS_FLOATS; // [48][16] patches p[k][n]

    // ---- stage zero-padded weights into LDS (whole workgroup) ----
    // Wlds[c][k] = weight[c*45 + k] for c<360 && k<45, else 0
    #pragma unroll 1
    for (int it = 0; it < WLDS_FLOATS / BLOCK; ++it) {     // exactly 69 full iterations
        int idx = tid + it * BLOCK;
        int c = idx / KPAD;
        int k = idx - c * KPAD;
        float v = 0.0f;
        if (c < ADIM * IKK && k < IKK) v = weight[c * IKK + k];
        wlds[idx] = v;
    }

    // ---- this wave's 16-pixel tile (16 consecutive x in one row) ----
    int g   = blockIdx.x * WAVES + wave;   // tile id
    int lin = g * TILE;                    // linear pixel base
    int b   = lin >> 14;                   // / (128*128)
    int rem = lin & 16383;
    int y   = rem >> 7;
    int x0  = rem & 127;

    // ---- build zero-padded patch tile p[k][n] in LDS (per wave) ----
    #pragma unroll 1
    for (int t = 0; t < PLDS_FLOATS / 32; ++t) {           // 24 iterations
        int idx = lane + 32 * t;
        int k = idx >> 4;
        int n = idx & 15;
        float v = 0.0f;
        if (k < IKK) {
            int i  = k / KSQ;
            int rr = k - i * KSQ;
            int di = rr / 3;
            int dj = rr - di * 3;
            int yy = y + di - 1;
            int xx = x0 + n + dj - 1;
            if (yy >= 0 && yy < HH && xx >= 0 && xx < WW) {
                v = (i < RDIM) ? rewards[((b * RDIM + i) * HH + yy) * WW + xx]
                               : values[(b * HH + yy) * WW + xx];
            }
        }
        plds[idx] = v;
    }

    __syncthreads();

    const int  n  = lane & 15;      // pixel within tile == WMMA N column == A row M
    const bool hi = lane >= 16;
    const int  kh = hi ? 2 : 0;     // K-half this lane supplies for A/B operands

    // ---- resident WMMA B operands: bs[s] = { p[4s+kh][n], p[4s+kh+1][n] } ----
    v2f bs[12];
    #pragma unroll
    for (int s = 0; s < 12; ++s) {
        int k0 = 4 * s + kh;
        bs[s][0] = plds[k0 * TILE + n];
        bs[s][1] = plds[(k0 + 1) * TILE + n];
    }

    // ---- resident patch values matching the C/D row ownership of this lane ----
    // D tile t, VGPR r: row j = 16t + r (+8 for hi lanes), column n = lane%16
    float pp[24];
    #pragma unroll
    for (int t = 0; t < 3; ++t)
        #pragma unroll
        for (int r = 0; r < 8; ++r) {
            int j = 16 * t + r + (hi ? 8 : 0);
            pp[t * 8 + r] = plds[j * TILE + n];
        }

    float best = -3.402823466e+38f;

    #pragma unroll 1
    for (int o = 0; o < ADIM; ++o) {
        v8f acc0 = {}, acc1 = {}, acc2 = {};
        const float* wb = wlds + (o * IKK) * KPAD + kh;    // row base for this action
        #pragma unroll
        for (int s = 0; s < 12; ++s) {
            // A operands: contiguous (k, k+1) f32 pair per lane -> ds_load_b64
            v2f a0 = *(const v2f*)(wb + (0 * 16 + n) * KPAD + 4 * s);
            v2f a1 = *(const v2f*)(wb + (1 * 16 + n) * KPAD + 4 * s);
            v2f a2 = *(const v2f*)(wb + (2 * 16 + n) * KPAD + 4 * s);
            acc0 = __builtin_amdgcn_wmma_f32_16x16x4_f32(false, a0, false, bs[s], (short)0, acc0, false, false);
            acc1 = __builtin_amdgcn_wmma_f32_16x16x4_f32(false, a1, false, bs[s], (short)0, acc1, false, false);
            acc2 = __builtin_amdgcn_wmma_f32_16x16x4_f32(false, a2, false, bs[s], (short)0, acc2, false, false);
        }

        // ---- softmax over j<45 + weighted sum with patch, split across lane halves ----
        float vmax = -3.402823466e+38f;
        #pragma unroll
        for (int r = 0; r < 8; ++r) vmax = fmaxf(vmax, acc0[r]);
        #pragma unroll
        for (int r = 0; r < 8; ++r) vmax = fmaxf(vmax, acc1[r]);
        #pragma unroll
        for (int r = 0; r < 8; ++r) {
            bool valid = (32 + r + (hi ? 8 : 0)) < IKK;    // mask padded j = 45..47
            vmax = fmaxf(vmax, valid ? acc2[r] : -3.402823466e+38f);
        }
        vmax = fmaxf(vmax, __shfl_xor(vmax, 16));

        float se = 0.0f, qe = 0.0f;
        #pragma unroll
        for (int r = 0; r < 8; ++r) {
            float e = __expf(acc0[r] - vmax);
            se += e; qe += e * pp[r];
        }
        #pragma unroll
        for (int r = 0; r < 8; ++r) {
            float e = __expf(acc1[r] - vmax);
            se += e; qe += e * pp[8 + r];
        }
        #pragma unroll
        for (int r = 0; r < 8; ++r) {
            bool valid = (32 + r + (hi ? 8 : 0)) < IKK;
            float e = valid ? __expf(acc2[r] - vmax) : 0.0f;
            se += e; qe += e * pp[16 + r];
        }
        se += __shfl_xor(se, 16);
        qe += __shfl_xor(qe, 16);
        best = fmaxf(best, qe / se);
    }

    if (!hi) out[lin + n] = best;
}

extern "C" void kernel_launch(void* const* d_in, const int* in_sizes, int n_in,
                              void* d_out, int out_size, void* d_ws, size_t ws_size,
                              hipStream_t stream) {
    (void)in_sizes; (void)n_in; (void)d_ws; (void)ws_size; (void)out_size;
    const float* values  = (const float*)d_in[0];
    const float* rewards = (const float*)d_in[1];
    const float* weight  = (const float*)d_in[2];
    float* out = (float*)d_out;

    const int tiles  = (BATCH * HH * WW) / TILE;   // 16384
    const int blocks = tiles / WAVES;              // 2048
    hipLaunchKernelGGL(vi_wmma_kernel, dim3(blocks), dim3(BLOCK), LDS_BYTES, stream,
                       values, rewards, weight, out);
}